// SparseLinear_31404800869166
// MI455X (gfx1250) — compile-verified
//
#include <hip/hip_runtime.h>
#include <stdint.h>
#include <stddef.h>

// Problem constants (match reference setup_inputs).
#define N_ROWS_C   100000
#define IN_F       1024
#define OUT_F      256
#define ROW_TILE   16
#define N_BLOCKS   (N_ROWS_C / ROW_TILE)   // 6250 exactly
#define XT_STRIDE  1028                    // 1024 + 4 pad -> conflict-free strided A reads
#define K_CHUNKS   (IN_F / 4)              // 256 chunks of K=4 for V_WMMA_F32_16X16X4_F32
#define STAGE_CAP  2048                    // CSR records staged in LDS per window

typedef __attribute__((ext_vector_type(2))) float v2f;
typedef __attribute__((ext_vector_type(8))) float v8f;

// ---------------------------------------------------------------------------
// 1) Repack weight [OUT_F x IN_F] into chunk-major Wp[kc][col][ki] so that the
//    two B values a lane needs (k = kc*4+kh, kc*4+kh+1 at one column) are
//    adjacent -> single coalesced global_load_b64 per N-tile per chunk.
// ---------------------------------------------------------------------------
__global__ void k_repack(const float* __restrict__ W, float* __restrict__ Wp) {
    const int idx = blockIdx.x * blockDim.x + threadIdx.x;   // exact grid: IN_F*OUT_F
    const int ki  = idx & 3;
    const int col = (idx >> 2) & (OUT_F - 1);
    const int kc  = idx >> 10;
    Wp[idx] = W[col * IN_F + (kc << 2) + ki];
}

// ---------------------------------------------------------------------------
// 2) Zero row-count array (ws is poisoned, must re-zero every call).
// ---------------------------------------------------------------------------
__global__ void k_zero(int* __restrict__ p, int n) {
    const int i = blockIdx.x * blockDim.x + threadIdx.x;
    if (i < n) p[i] = 0;
}

// ---------------------------------------------------------------------------
// 3) Histogram of nnz per row.
// ---------------------------------------------------------------------------
__global__ void k_hist(const int* __restrict__ rows, int* __restrict__ cnt, int nnz) {
    const int i = blockIdx.x * blockDim.x + threadIdx.x;
    if (i < nnz) atomicAdd(&cnt[rows[i]], 1);
}

// ---------------------------------------------------------------------------
// 4) Exclusive scan of counts (single 1024-thread block, chunked with carry).
//    Also seeds the scatter cursor array.
// ---------------------------------------------------------------------------
__global__ void k_scan(const int* __restrict__ cnt, int* __restrict__ roff,
                       int* __restrict__ cur, int n) {
    __shared__ int buf[1024];
    __shared__ int carry;
    const int tid = threadIdx.x;
    if (tid == 0) carry = 0;
    __syncthreads();
    for (int base = 0; base < n; base += 1024) {
        const int i = base + tid;
        const int v = (i < n) ? cnt[i] : 0;
        buf[tid] = v;
        __syncthreads();
        for (int s = 1; s < 1024; s <<= 1) {
            const int t = (tid >= s) ? buf[tid - s] : 0;
            __syncthreads();
            buf[tid] += t;
            __syncthreads();
        }
        if (i < n) {
            const int ex = carry + buf[tid] - v;   // exclusive prefix
            roff[i] = ex;
            cur[i]  = ex;
        }
        __syncthreads();                            // all carry reads done
        if (tid == 1023) carry += buf[1023];
        __syncthreads();
    }
    if (tid == 0) roff[n] = carry;                  // == nnz
}

// ---------------------------------------------------------------------------
// 5) Scatter COO entries into CSR order as packed 8-byte (col,val) records
//    (duplicates kept; densify sums them). One b64 store per entry.
// ---------------------------------------------------------------------------
__global__ void k_scatter(const int* __restrict__ rows, const int* __restrict__ cols,
                          const float* __restrict__ vals, int* __restrict__ cur,
                          unsigned long long* __restrict__ cpair, int nnz) {
    const int i = blockIdx.x * blockDim.x + threadIdx.x;
    if (i < nnz) {
        const int p = atomicAdd(&cur[rows[i]], 1);
        cpair[p] = (unsigned long long)(unsigned)cols[i] |
                   ((unsigned long long)__float_as_uint(vals[i]) << 32);
    }
}

// ---------------------------------------------------------------------------
// 6) Main kernel: per 16-row block
//    - async-copy the block's contiguous CSR slice into LDS (ASYNCcnt path),
//      overlapped with b128 zero-fill of the dense tile
//    - densify + build 256-bit K-chunk occupancy mask
//    - ballot-compact occupied chunks into an ordered list
//    - software-pipelined dense f32 GEMM on the WMMA pipe over occupied chunks
// ---------------------------------------------------------------------------
__global__ __launch_bounds__(128) void k_spmm_wmma(
    const int* __restrict__ roff, const unsigned long long* __restrict__ cpair,
    const float* __restrict__ Wp, const float* __restrict__ bias,
    float* __restrict__ out)
{
    __shared__ __align__(16) float xt[ROW_TILE * XT_STRIDE];   // 16x1024(+pad) f32 tile
    __shared__ unsigned long long  spair[STAGE_CAP];           // staged CSR records
    __shared__ unsigned            cmask[8];                   // 256-bit chunk occupancy
    __shared__ unsigned short      clist[K_CHUNKS];            // compacted chunk indices
    __shared__ int                 n_occ;

    const int tid  = threadIdx.x;
    const int lane = tid & 31;
    const int wave = tid >> 5;
    const int row0 = blockIdx.x * ROW_TILE;

    // Block's CSR slice bounds (uniform across the block).
    const int s0 = roff[row0];
    const int e0 = roff[row0 + ROW_TILE];

    // Issue the async LDS copy of the first window, then overlap the tile
    // zero-fill with it (disjoint LDS regions).
    int base = s0;
    int wcnt = min(e0 - base, STAGE_CAP);
    for (int j = tid; j < wcnt; j += 128) {
        const unsigned           ldsa = (unsigned)(uintptr_t)(const void*)&spair[j];
        const unsigned long long gva  = (unsigned long long)(uintptr_t)(cpair + base + j);
        asm volatile("global_load_async_to_lds_b64 %0, %1, off"
                     :: "v"(ldsa), "v"(gva) : "memory");
    }
    {   // zero dense tile with ds_store_b128
        float4* xt4 = (float4*)xt;
        const float4 z = {0.0f, 0.0f, 0.0f, 0.0f};
        for (int i = tid; i < (ROW_TILE * XT_STRIDE) / 4; i += 128) xt4[i] = z;
        if (tid < 8) cmask[tid] = 0u;
    }

    // Densify window-by-window (one window in the common case: ~160 records).
    while (true) {
        asm volatile("s_wait_asynccnt 0x0" ::: "memory");
        __syncthreads();
        if (tid < ROW_TILE) {
            // Lane r owns row r: serial RMW is race-free and sums duplicates.
            const int gs = roff[row0 + tid];
            const int ge = roff[row0 + tid + 1];
            const int js = max(gs, base) - base;
            const int je = min(ge, base + wcnt) - base;
            float* xrow = &xt[tid * XT_STRIDE];
            for (int j = js; j < je; ++j) {
                const unsigned long long u = spair[j];
                const int c = (int)(u & 0xffffffffu);
                xrow[c] += __uint_as_float((unsigned)(u >> 32));
                atomicOr(&cmask[c >> 7], 1u << ((c >> 2) & 31));
            }
        }
        __syncthreads();
        base += wcnt;
        if (base >= e0) break;
        wcnt = min(e0 - base, STAGE_CAP);
        for (int j = tid; j < wcnt; j += 128) {
            const unsigned           ldsa = (unsigned)(uintptr_t)(const void*)&spair[j];
            const unsigned long long gva  = (unsigned long long)(uintptr_t)(cpair + base + j);
            asm volatile("global_load_async_to_lds_b64 %0, %1, off"
                         :: "v"(ldsa), "v"(gva) : "memory");
        }
    }

    // Compact occupied chunk indices (ordered) via wave-0 ballot prefix.
    if (wave == 0) {
        int bc = 0;
        #pragma unroll
        for (int g = 0; g < 8; ++g) {
            const unsigned w = cmask[g];
            if ((w >> lane) & 1u)
                clist[bc + __popc(w & ((1u << lane) - 1u))] =
                    (unsigned short)((g << 5) | lane);
            bc += __popc(w);
        }
        if (lane == 0) n_occ = bc;
    }
    __syncthreads();

    // ---- WMMA main loop -----------------------------------------------------
    const int m    = lane & 15;                 // A-matrix M row for this lane
    const int kh   = (lane >> 4) << 1;          // K sub-offset: 0 (lanes 0-15) / 2
    const int ncol = (wave << 6) | (lane & 15); // base output column of tile 0

    v8f acc[4];
    #pragma unroll
    for (int t = 0; t < 4; ++t) acc[t] = (v8f){0,0,0,0,0,0,0,0};

    // Fetch helper: A from LDS (one ds_load_b64), 4 B tiles as coalesced b64.
    auto fetch = [&](int kc, v2f& a, v2f& q0, v2f& q1, v2f& q2, v2f& q3) {
        const int k = (kc << 2) + kh;
        a.x = xt[m * XT_STRIDE + k];
        a.y = xt[m * XT_STRIDE + k + 1];
        const float* wp = Wp + (kc << 10) + (ncol << 2) + kh;
        q0 = *(const v2f*)(wp);
        q1 = *(const v2f*)(wp + 64);
        q2 = *(const v2f*)(wp + 128);
        q3 = *(const v2f*)(wp + 192);
    };

    const int nocc = n_occ;                     // uniform
    if (nocc > 0) {
        v2f a, b0, b1, b2, b3;
        fetch(clist[0], a, b0, b1, b2, b3);
        // Software pipeline: next chunk's loads fly over current chunk's WMMAs.
        for (int i = 1; i < nocc; ++i) {
            v2f an, c0, c1, c2, c3;
            fetch(clist[i], an, c0, c1, c2, c3);
            acc[0] = __builtin_amdgcn_wmma_f32_16x16x4_f32(false, a, false, b0, (short)0, acc[0], false, false);
            acc[1] = __builtin_amdgcn_wmma_f32_16x16x4_f32(false, a, false, b1, (short)0, acc[1], false, false);
            acc[2] = __builtin_amdgcn_wmma_f32_16x16x4_f32(false, a, false, b2, (short)0, acc[2], false, false);
            acc[3] = __builtin_amdgcn_wmma_f32_16x16x4_f32(false, a, false, b3, (short)0, acc[3], false, false);
            a = an; b0 = c0; b1 = c1; b2 = c2; b3 = c3;
        }
        acc[0] = __builtin_amdgcn_wmma_f32_16x16x4_f32(false, a, false, b0, (short)0, acc[0], false, false);
        acc[1] = __builtin_amdgcn_wmma_f32_16x16x4_f32(false, a, false, b1, (short)0, acc[1], false, false);
        acc[2] = __builtin_amdgcn_wmma_f32_16x16x4_f32(false, a, false, b2, (short)0, acc[2], false, false);
        acc[3] = __builtin_amdgcn_wmma_f32_16x16x4_f32(false, a, false, b3, (short)0, acc[3], false, false);
    }

    // Epilogue: C element i of acc is (M = i + 8*(lane>=16), N = lane&15).
    const int mbase = (lane >> 4) << 3;
    #pragma unroll
    for (int t = 0; t < 4; ++t) {
        const int col = ncol + t * 16;
        const float bv = bias[col];
        #pragma unroll
        for (int i = 0; i < 8; ++i) {
            const int r = row0 + mbase + i;
            __builtin_nontemporal_store(acc[t][i] + bv, &out[(size_t)r * OUT_F + col]);
        }
    }
}

// ---------------------------------------------------------------------------
// Launch pipeline (all on `stream`, graph-capture safe: no mallocs/syncs).
// ---------------------------------------------------------------------------
extern "C" void kernel_launch(void* const* d_in, const int* in_sizes, int n_in,
                              void* d_out, int out_size, void* d_ws, size_t ws_size,
                              hipStream_t stream)
{
    const int*   rows = (const int*)  d_in[0];
    const int*   cols = (const int*)  d_in[1];
    const float* vals = (const float*)d_in[2];
    // d_in[3] = n_rows scalar (compile-time constant here)
    const float* W    = (const float*)d_in[4];
    const float* bias = (const float*)d_in[5];
    const int    nnz  = in_sizes[0];
    float*       out  = (float*)d_out;
    (void)n_in; (void)out_size; (void)ws_size;

    // Carve workspace (~10.8 MB), 256B-aligned slices.
    uint8_t* ws = (uint8_t*)d_ws;
    size_t pos = 0;
    auto carve = [&](size_t bytes) -> void* {
        void* p = ws + pos;
        pos = (pos + bytes + 255) & ~(size_t)255;
        return p;
    };
    float*              Wp    = (float*)carve(sizeof(float) * IN_F * OUT_F);
    int*                cnt   = (int*)  carve(sizeof(int)   * N_ROWS_C);
    int*                roff  = (int*)  carve(sizeof(int)   * (N_ROWS_C + 1));
    int*                cur   = (int*)  carve(sizeof(int)   * N_ROWS_C);
    unsigned long long* cpair = (unsigned long long*)carve(sizeof(unsigned long long) * nnz);

    hipLaunchKernelGGL(k_repack,    dim3((IN_F * OUT_F) / 256), dim3(256), 0, stream, W, Wp);
    hipLaunchKernelGGL(k_zero,      dim3((N_ROWS_C + 255) / 256), dim3(256), 0, stream, cnt, N_ROWS_C);
    hipLaunchKernelGGL(k_hist,      dim3((nnz + 255) / 256), dim3(256), 0, stream, rows, cnt, nnz);
    hipLaunchKernelGGL(k_scan,      dim3(1), dim3(1024), 0, stream, cnt, roff, cur, N_ROWS_C);
    hipLaunchKernelGGL(k_scatter,   dim3((nnz + 255) / 256), dim3(256), 0, stream,
                       rows, cols, vals, cur, cpair, nnz);
    hipLaunchKernelGGL(k_spmm_wmma, dim3(N_BLOCKS), dim3(128), 0, stream,
                       roff, cpair, Wp, bias, out);
}